// UnsupervisedLoss_6262062318054
// MI455X (gfx1250) — compile-verified
//
#include <hip/hip_runtime.h>
#include <math.h>

// ---------------------------------------------------------------------------
// Problem constants (match the JAX reference)
// ---------------------------------------------------------------------------
#define BB 4
#define NN 2048
#define MM 2048
#define KNN 64
#define EMD_ITERS 50
// eps = 0.005 -> logK = -C/eps = C * (-200)
#define LOGK_SCALE (-200.0f)

typedef __attribute__((ext_vector_type(2))) float v2f;
typedef __attribute__((ext_vector_type(8))) float v8f;

// ---------------------------------------------------------------------------
// Workspace layout (float offsets)
// ---------------------------------------------------------------------------
#define OFF_MAT    ((size_t)0)                          // 16777216 f  (64MB): logK, later d2(inputs,inputs)
#define OFF_LU     ((size_t)16777216)                   // 8192 f
#define OFF_LV     (OFF_LU + 8192)                      // 8192 f
#define OFF_NA     (OFF_LV + 8192)                      // 8192 f  |inputs|^2
#define OFF_NB     (OFF_NA + 8192)                      // 8192 f  |preds|^2
#define OFF_PM     (OFF_NB + 8192)                      // 131072 f partial max
#define OFF_PS     (OFF_PM + 131072)                    // 131072 f partial sum
#define OFF_QLOSS  (OFF_PS + 131072)                    // 8192 f
#define OFF_ASSIGN (OFF_QLOSS + 8192)                   // 8192 i
#define OFF_KIDX   (OFF_ASSIGN + 8192)                  // 524288 i
#define OFF_KDIST  (OFF_KIDX + 524288)                  // 524288 f

// Branchless single-exp online logsumexp update.
__device__ __forceinline__ void lse_accum(float& m, float& s, float x) {
    float mn = fmaxf(m, x);
    float e  = __expf(fminf(m, x) - mn);     // exp(smaller - larger), exp(-inf)=0 on first elem
    s = (x > m) ? fmaf(s, e, 1.0f) : (s + e);
    m = mn;
}
// Merge two (max,sum) pairs.
__device__ __forceinline__ void lse_merge(float& m, float& s, float mo, float so) {
    float mn = fmaxf(m, mo);
    s = s * __expf(m - mn) + so * __expf(mo - mn);
    m = mn;
}

// ---------------------------------------------------------------------------
// 0. point norms
// ---------------------------------------------------------------------------
__global__ void norms_kernel(const float* __restrict__ pa, const float* __restrict__ pb,
                             float* __restrict__ na, float* __restrict__ nb) {
    int t = blockIdx.x * 256 + threadIdx.x;
    if (t < BB * NN) {
        const float* p = pa + (size_t)t * 3;
        na[t] = p[0]*p[0] + p[1]*p[1] + p[2]*p[2];
        const float* q = pb + (size_t)t * 3;
        nb[t] = q[0]*q[0] + q[1]*q[1] + q[2]*q[2];
    }
}

// ---------------------------------------------------------------------------
// 1. pairwise squared distance via V_WMMA_F32_16X16X4_F32
//    out[b][i][j] = scale * max(|a_i|^2 + |b_j|^2 - 2 a_i.b_j, 0)
//    One wave per 16x16 tile. A fragment: 16x4 (K padded: x,y,z,0).
//    Lane layout (ISA 7.12.2): lanes 0-15 hold K=0,1; lanes 16-31 hold K=2,3.
// ---------------------------------------------------------------------------
__global__ void pairdist_wmma_kernel(const float* __restrict__ Ap, const float* __restrict__ Bp,
                                     const float* __restrict__ na, const float* __restrict__ nb,
                                     float* __restrict__ out, float scale) {
    const int tid  = threadIdx.x;
    const int lane = tid & 31;
    const int wave = tid >> 5;
    int tile = blockIdx.x * 8 + wave;          // 65536 tiles total
    int b  = tile >> 14;                       // 16384 tiles per batch
    int r  = tile & 16383;
    int i0 = (r >> 7) << 4;
    int j0 = (r & 127) << 4;

    int m  = lane & 15;
    int ks = lane >> 4;                        // 0: K=0,1   1: K=2,3

    const float* ap = Ap + ((size_t)b * NN + i0 + m) * 3;
    const float* bp = Bp + ((size_t)b * MM + j0 + m) * 3;
    float ax = ap[0], ay = ap[1], az = ap[2];
    float bx = bp[0], by = bp[1], bz = bp[2];

    v2f a, bbf;
    a.x   = ks ? az : ax;  a.y   = ks ? 0.0f : ay;
    bbf.x = ks ? bz : bx;  bbf.y = ks ? 0.0f : by;

    v8f c = {};
    // D = A x B (+0): 16x16 tile of dot products
    v8f d = __builtin_amdgcn_wmma_f32_16x16x4_f32(false, a, false, bbf,
                                                  (short)0, c, false, false);

    float nbj = nb[(size_t)b * MM + j0 + m];                 // column norm (col == lane&15)
    size_t colbase = ((size_t)b * NN + i0) * MM + j0 + m;
#pragma unroll
    for (int r8 = 0; r8 < 8; ++r8) {
        int row = r8 + ks * 8;                               // D layout: VGPR r -> M=r / M=r+8
        float nai = na[(size_t)b * NN + i0 + row];
        float dd  = nai + nbj - 2.0f * d[r8];
        dd = fmaxf(dd, 0.0f);
        out[colbase + (size_t)row * MM] = dd * scale;
    }
}

// ---------------------------------------------------------------------------
// 2. Sinkhorn in log domain
// ---------------------------------------------------------------------------
__global__ void zero_uv_kernel(float* __restrict__ lu, float* __restrict__ lv) {
    int t = blockIdx.x * 256 + threadIdx.x;
    if (t < BB * NN) { lu[t] = 0.0f; lv[t] = 0.0f; }
}

// lu[b,i] = lmu - LSE_j( logK[b,i,j] + lv[b,j] )
// One wave per row; float4 loads (512B per wave per clause).
__global__ void row_lse_kernel(const float* __restrict__ mat, const float* __restrict__ lv,
                               float* __restrict__ lu, float lmu) {
    int r    = blockIdx.x * 8 + (threadIdx.x >> 5);          // global row over B*N
    int lane = threadIdx.x & 31;
    int b    = r >> 11;
    const float4* row = (const float4*)(mat + (size_t)r * MM);
    const float4* lvb = (const float4*)(lv + (size_t)b * MM);
    float m = -INFINITY, s = 0.0f;
    for (int jj = lane; jj < MM / 4; jj += 32) {             // 16 iterations
        float4 v = row[jj];
        float4 w = lvb[jj];
        lse_accum(m, s, v.x + w.x);
        lse_accum(m, s, v.y + w.y);
        lse_accum(m, s, v.z + w.z);
        lse_accum(m, s, v.w + w.w);
    }
#pragma unroll
    for (int off = 16; off; off >>= 1) {
        float mo = __shfl_xor(m, off, 32);
        float so = __shfl_xor(s, off, 32);
        lse_merge(m, s, mo, so);
    }
    if (lane == 0) lu[r] = lmu - (m + __logf(s));
}

// stage A of column LSE: 16 row-segments of 128 rows.
// Each thread owns 4 adjacent columns (float4 loads, coalesced).
__global__ void col_partial_kernel(const float* __restrict__ mat, const float* __restrict__ lu,
                                   float* __restrict__ pm, float* __restrict__ ps) {
    int j0  = (blockIdx.x * 256 + threadIdx.x) * 4;   // 0..2044, gridDim.x = 2
    int seg = blockIdx.y;                              // 0..15
    int b   = blockIdx.z;                              // 0..3
    const float* lub  = lu + (size_t)b * NN + seg * 128;
    const float* base = mat + ((size_t)b * NN + seg * 128) * MM + j0;
    float m[4] = {-INFINITY, -INFINITY, -INFINITY, -INFINITY};
    float s[4] = {0.0f, 0.0f, 0.0f, 0.0f};
    for (int i = 0; i < 128; ++i) {
        float4 v = *(const float4*)(base + (size_t)i * MM);
        float li = lub[i];
        lse_accum(m[0], s[0], v.x + li);
        lse_accum(m[1], s[1], v.y + li);
        lse_accum(m[2], s[2], v.z + li);
        lse_accum(m[3], s[3], v.w + li);
    }
    size_t o = ((size_t)(b * 16 + seg)) * MM + j0;
    *(float4*)(pm + o) = make_float4(m[0], m[1], m[2], m[3]);
    *(float4*)(ps + o) = make_float4(s[0], s[1], s[2], s[3]);
}

// stage B: merge 16 (max,sum) partials -> lv[b,j] = lnu - LSE   (4 cols/thread)
__global__ void col_combine_kernel(const float* __restrict__ pm, const float* __restrict__ ps,
                                   float* __restrict__ lv, float lnu) {
    int t0 = (blockIdx.x * 256 + threadIdx.x) * 4;    // 0..8188, gridDim.x = 8
    int b = t0 >> 11; int j = t0 & (MM - 1);
    float m[4] = {-INFINITY, -INFINITY, -INFINITY, -INFINITY};
    float s[4] = {0.0f, 0.0f, 0.0f, 0.0f};
#pragma unroll
    for (int seg = 0; seg < 16; ++seg) {
        size_t o = ((size_t)(b * 16 + seg)) * MM + j;
        float4 mo = *(const float4*)(pm + o);
        float4 so = *(const float4*)(ps + o);
        lse_merge(m[0], s[0], mo.x, so.x);
        lse_merge(m[1], s[1], mo.y, so.y);
        lse_merge(m[2], s[2], mo.z, so.z);
        lse_merge(m[3], s[3], mo.w, so.w);
    }
    float4 r;
    r.x = lnu - (m[0] + __logf(s[0]));
    r.y = lnu - (m[1] + __logf(s[1]));
    r.z = lnu - (m[2] + __logf(s[2]));
    r.w = lnu - (m[3] + __logf(s[3]));
    *(float4*)(lv + t0) = r;
}

// argmax_j of (logK + lu_i + lv_j): lu_i is constant per row -> drop it.
// Tie-break: lowest j (matches jnp.argmax).  float4 loads.
__global__ void argmax_kernel(const float* __restrict__ mat, const float* __restrict__ lv,
                              int* __restrict__ assign) {
    int r    = blockIdx.x * 8 + (threadIdx.x >> 5);
    int lane = threadIdx.x & 31;
    int b    = r >> 11;
    const float4* row = (const float4*)(mat + (size_t)r * MM);
    const float4* lvb = (const float4*)(lv + (size_t)b * MM);
    float best = -INFINITY; int bidx = 0x7fffffff;
    for (int jj = lane; jj < MM / 4; jj += 32) {
        float4 v = row[jj];
        float4 w = lvb[jj];
        float x0 = v.x + w.x, x1 = v.y + w.y, x2 = v.z + w.z, x3 = v.w + w.w;
        int j = jj * 4;
        if (x0 > best) { best = x0; bidx = j; }
        if (x1 > best) { best = x1; bidx = j + 1; }
        if (x2 > best) { best = x2; bidx = j + 2; }
        if (x3 > best) { best = x3; bidx = j + 3; }
    }
#pragma unroll
    for (int off = 16; off; off >>= 1) {
        float ov = __shfl_xor(best, off, 32);
        int   oi = __shfl_xor(bidx, off, 32);
        if (ov > best || (ov == best && oi < bidx)) { best = ov; bidx = oi; }
    }
    if (lane == 0) assign[r] = bidx;
}

// ---------------------------------------------------------------------------
// 3. kNN: top-65 smallest per row of d2(inputs,inputs); drop entry 0 (self).
//    Stable insertion (strict <) reproduces top_k's lowest-index tie-break.
// ---------------------------------------------------------------------------
__global__ void knn_select_kernel(const float* __restrict__ d2,
                                  int* __restrict__ kidx, float* __restrict__ kdist) {
    int t = blockIdx.x * 256 + threadIdx.x;     // query over B*N
    if (t >= BB * NN) return;
    const float4* row = (const float4*)(d2 + (size_t)t * MM);
    float bd[KNN + 1]; int bi[KNN + 1];
#pragma unroll
    for (int k = 0; k <= KNN; ++k) { bd[k] = INFINITY; bi[k] = -1; }
    for (int jj = 0; jj < MM / 4; ++jj) {
        float4 v = row[jj];
        float c[4] = {v.x, v.y, v.z, v.w};
#pragma unroll
        for (int u = 0; u < 4; ++u) {
            float d = c[u];
            if (d < bd[KNN]) {
                int j = jj * 4 + u;
                int p = KNN;
                while (p > 0 && d < bd[p - 1]) { bd[p] = bd[p - 1]; bi[p] = bi[p - 1]; --p; }
                bd[p] = d; bi[p] = j;
            }
        }
    }
    for (int k = 0; k < KNN; ++k) {
        kidx[(size_t)t * KNN + k]  = bi[k + 1];
        kdist[(size_t)t * KNN + k] = bd[k + 1];
    }
}

// ---------------------------------------------------------------------------
// 4. JAX threefry2x32 (key(42) -> k=[0,42]); counter split-in-half layout.
// ---------------------------------------------------------------------------
__device__ __forceinline__ unsigned rotl32(unsigned x, int d) { return (x << d) | (x >> (32 - d)); }

__device__ float jax_uniform(unsigned e) {
    const unsigned H = (BB * NN * KNN) / 2;      // 262144
    unsigned p = (e < H) ? e : e - H;
    unsigned x0 = p, x1 = p + H;
    const unsigned k0 = 0u, k1 = 42u;
    const unsigned ks[3] = {k0, k1, k0 ^ k1 ^ 0x1BD11BDAu};
    const int rotA[4] = {13, 15, 26, 6};
    const int rotB[4] = {17, 29, 16, 24};
    x0 += ks[0]; x1 += ks[1];
#pragma unroll
    for (int g = 0; g < 5; ++g) {
#pragma unroll
        for (int rr = 0; rr < 4; ++rr) {
            int rot = (g & 1) ? rotB[rr] : rotA[rr];
            x0 += x1; x1 = rotl32(x1, rot); x1 ^= x0;
        }
        x0 += ks[(g + 1) % 3];
        x1 += ks[(g + 2) % 3] + (unsigned)(g + 1);
    }
    unsigned bits = (e < H) ? x0 : x1;
    unsigned fb = (bits >> 9) | 0x3f800000u;
    return __uint_as_float(fb) - 1.0f;
}

// ---------------------------------------------------------------------------
// 5. per-query masked average distance (one wave per query, 2 k's per lane)
// ---------------------------------------------------------------------------
__global__ void qloss_kernel(const float* __restrict__ preds, const float* __restrict__ inputs,
                             const int* __restrict__ assign, const int* __restrict__ kidx,
                             const float* __restrict__ kdist, float* __restrict__ qloss) {
    int q    = blockIdx.x * 8 + (threadIdx.x >> 5);
    int lane = threadIdx.x & 31;
    int b    = q >> 11;
    int a    = assign[q];
    const float* pp = preds + ((size_t)b * NN + a) * 3;
    float px = pp[0], py = pp[1], pz = pp[2];

    float msum = 0.0f, wsum = 0.0f, d63 = 0.0f;
#pragma unroll
    for (int h = 0; h < 2; ++h) {
        int k = lane + 32 * h;
        int o = q * KNN + k;
        int j = kidx[o];
        float kd = kdist[o];
        float prob = expf(-(kd / (0.05f * 0.05f)) / (2.0f * 0.5657f * 0.5657f))
                     / (0.5657f * 2.5066282746f);
        float u  = jax_uniform((unsigned)o);
        float mk = (u < prob) ? 1.0f : 0.0f;
        const float* nbp = inputs + ((size_t)b * NN + j) * 3;
        float dx = px - nbp[0], dy = py - nbp[1], dz = pz - nbp[2];
        float dd = dx * dx + dy * dy + dz * dz;
        msum += mk; wsum += mk * dd;
        if (k == KNN - 1) d63 = dd;
    }
#pragma unroll
    for (int off = 16; off; off >>= 1) {
        msum += __shfl_xor(msum, off, 32);
        wsum += __shfl_xor(wsum, off, 32);
    }
    d63 = __shfl(d63, 31, 32);                   // fallback when no neighbor sampled
    float avg = (msum == 0.0f) ? d63 : (wsum / msum);
    if (lane == 0) qloss[q] = avg;
}

// deterministic final sum (fixed order -> bit-stable across graph replays)
__global__ void final_reduce_kernel(const float* __restrict__ qloss, float* __restrict__ out) {
    __shared__ float sm[256];
    int tid = threadIdx.x;
    float s = 0.0f;
    for (int i = tid; i < BB * NN; i += 256) s += qloss[i];
    sm[tid] = s; __syncthreads();
    for (int st = 128; st; st >>= 1) { if (tid < st) sm[tid] += sm[tid + st]; __syncthreads(); }
    if (tid == 0) out[0] = sm[0];
}

// ---------------------------------------------------------------------------
extern "C" void kernel_launch(void* const* d_in, const int* in_sizes, int n_in,
                              void* d_out, int out_size, void* d_ws, size_t ws_size,
                              hipStream_t stream) {
    const float* preds  = (const float*)d_in[0];
    const float* inputs = (const float*)d_in[1];
    (void)in_sizes; (void)n_in; (void)out_size; (void)ws_size;

    float* W      = (float*)d_ws;
    float* mat    = W + OFF_MAT;
    float* lu     = W + OFF_LU;
    float* lv     = W + OFF_LV;
    float* na     = W + OFF_NA;
    float* nb     = W + OFF_NB;
    float* pm     = W + OFF_PM;
    float* ps     = W + OFF_PS;
    float* qloss  = W + OFF_QLOSS;
    int*   assign = (int*)(W + OFF_ASSIGN);
    int*   kidx   = (int*)(W + OFF_KIDX);
    float* kdist  = W + OFF_KDIST;

    const float lmu = -logf((float)NN);
    const float lnu = -logf((float)MM);

    // norms of inputs (A of EMD cost) and preds (B of EMD cost)
    norms_kernel<<<32, 256, 0, stream>>>(inputs, preds, na, nb);

    // logK = -C/eps via WMMA (4*128*128 tiles / 8 waves per block)
    pairdist_wmma_kernel<<<8192, 256, 0, stream>>>(inputs, preds, na, nb, mat, LOGK_SCALE);

    zero_uv_kernel<<<32, 256, 0, stream>>>(lu, lv);

    for (int it = 0; it < EMD_ITERS; ++it) {
        row_lse_kernel<<<1024, 256, 0, stream>>>(mat, lv, lu, lmu);
        col_partial_kernel<<<dim3(2, 16, BB), 256, 0, stream>>>(mat, lu, pm, ps);
        col_combine_kernel<<<8, 256, 0, stream>>>(pm, ps, lv, lnu);
    }
    argmax_kernel<<<1024, 256, 0, stream>>>(mat, lv, assign);

    // reuse mat for d2(inputs, inputs) (WMMA again), then kNN top-65
    pairdist_wmma_kernel<<<8192, 256, 0, stream>>>(inputs, inputs, na, na, mat, 1.0f);
    knn_select_kernel<<<32, 256, 0, stream>>>(mat, kidx, kdist);

    qloss_kernel<<<1024, 256, 0, stream>>>(preds, inputs, assign, kidx, kdist, qloss);
    final_reduce_kernel<<<1, 256, 0, stream>>>(qloss, (float*)d_out);
}